// DeepAggregateLayer_11149735100495
// MI455X (gfx1250) — compile-verified
//
#include <hip/hip_runtime.h>

// DeepAggregateLayer: out[o] = (op[o]==0 ? min : max) over x[conn[o, 0..127]]
//
// MI455X / gfx1250 strategy:
//  - HBM floor = 134 MB of streaming index reads (~5.8 us @ 23.3 TB/s).
//  - x (256 KB) staged into LDS once per workgroup via the Tensor Data Mover
//    (tensor_load_to_lds + s_wait_tensorcnt): 33.5M random 4B gathers then
//    hit LDS banks instead of serializing the VMEM pipe.
//  - One wave32 per row: coalesced global_load_b128 index loads, 4 LDS
//    gathers/lane, then a split-half reduction (lanes 0-15 reduce min,
//    lanes 16-31 reduce max) using 5 ds_swizzle_b32 xor steps total for
//    BOTH reductions (vs 10 bpermutes + VALU index setup for __shfl_xor).

typedef __attribute__((ext_vector_type(4))) unsigned int u32x4;
typedef __attribute__((ext_vector_type(8))) int          i32x8;
typedef __attribute__((ext_vector_type(4))) int          i32x4;

#define IN_FEATURES   65536
#define OUT_FEATURES  262144
#define NUM_CONN      128
#define THREADS       512
#define WAVES         (THREADS / 32)
#define ROWS_PER_BLK  256
#define NBLOCKS       (OUT_FEATURES / ROWS_PER_BLK)

// ds_swizzle group-of-32 xor pattern: offset = (xor_mask<<10) | (or=0) | (and=0x1f)
#define SWZ_XOR(m) (((m) << 10) | 0x1f)

__device__ __forceinline__ float swz_xor_f(float v, const int imm_done_below) {
  return v; // (unused helper placeholder; real calls are macro-expanded below)
}

#define DS_SWZ_F(v, mask)                                                     \
  __int_as_float(__builtin_amdgcn_ds_swizzle(__float_as_int(v), SWZ_XOR(mask)))

__global__ __launch_bounds__(THREADS, 1)
void deep_aggregate_kernel(const float* __restrict__ x,
                           const int*   __restrict__ conn,
                           const int*   __restrict__ ops,
                           float*       __restrict__ out) {
  // Dynamic LDS: 65536 floats = 256 KB, at LDS offset 0 (only allocation).
  extern __shared__ float lds_x[];

  const int lane = threadIdx.x & 31;
  const int wave = threadIdx.x >> 5;

  // ---- Stage x into LDS with one TDM descriptor (issued by wave 0) ----
  if (wave == 0) {
    unsigned long long ga = (unsigned long long)(size_t)x;

    // D# group 0 (128b): count=1 | lds_addr=0 | global_addr[56:0] | type=2
    u32x4 g0;
    g0[0] = 1u;                                        // count=1, is_restore=0
    g0[1] = 0u;                                        // lds_addr = 0 bytes
    g0[2] = (unsigned)(ga & 0xFFFFFFFFull);            // global_addr[31:0]
    g0[3] = (unsigned)((ga >> 32) & 0x1FFFFFFull)      // global_addr[56:32]
          | (2u << 30);                                // type = 2 ("image")

    // D# group 1 (256b): 2D tensor 8192 x 8 of 4B elems, tile = whole tensor.
    i32x8 g1;
    g1[0] = (int)(2u << 16);        // workgroup_mask=0, data_size=2 (4 bytes)
    g1[1] = (int)(8192u << 16);     // atomic_barrier_addr=0 | tensor_dim0.lo16
    g1[2] = (int)(8u << 16);        // tensor_dim0.hi16=0    | tensor_dim1.lo16
    g1[3] = (int)(8192u << 16);     // tensor_dim1.hi16=0    | tile_dim0=8192
    g1[4] = 8;                      // tile_dim1=8, tile_dim2=0
    g1[5] = 8192;                   // tensor_dim0_stride[31:0] = 8192
    g1[6] = 0;                      // stride0.hi16=0, dim1_stride.lo16=0
    g1[7] = 0;

    i32x4 gz4 = {0, 0, 0, 0};                 // groups 2/3 unused (2-D tensor)
    i32x8 gz8 = {0, 0, 0, 0, 0, 0, 0, 0};     // extra group (clang-23 form)

    __builtin_amdgcn_tensor_load_to_lds(g0, g1, gz4, gz4, gz8, 0);
    __builtin_amdgcn_s_wait_tensorcnt(0);
  }
  __syncthreads();

  const bool low_half = (lane < 16);   // lanes 0-15: min tree; 16-31: max tree

  // ---- One wave per output row ----
  const int row_base = blockIdx.x * ROWS_PER_BLK;
  #pragma unroll 2
  for (int r = wave; r < ROWS_PER_BLK; r += WAVES) {
    const int row = row_base + r;

    // 128 indices / row; lane takes 4 -> one coalesced global_load_b128/lane.
    const i32x4* idx4 = (const i32x4*)(conn + (size_t)row * NUM_CONN);
    i32x4 c = idx4[lane];

    // Hoist the operator load (uniform broadcast) to overlap with the gathers.
    const int op = ops[row];

    float v0 = lds_x[c[0]];
    float v1 = lds_x[c[1]];
    float v2 = lds_x[c[2]];
    float v3 = lds_x[c[3]];

    float mn = fminf(fminf(v0, v1), fminf(v2, v3));
    float mx = fmaxf(fmaxf(v0, v1), fmaxf(v2, v3));

    // Cross-half seed: low lanes send their max up, high lanes send their
    // min down; afterwards low lanes hold pairwise mins, high lanes maxs.
    {
      float send = low_half ? mx : mn;
      float recv = DS_SWZ_F(send, 16);
      mn = low_half ? fminf(mn, recv) : fmaxf(mx, recv);  // 'mn' = my tree value
    }
    // 4-step butterfly within each half reduces both trees at once.
    {
      float t;
      t = DS_SWZ_F(mn, 8); mn = low_half ? fminf(mn, t) : fmaxf(mn, t);
      t = DS_SWZ_F(mn, 4); mn = low_half ? fminf(mn, t) : fmaxf(mn, t);
      t = DS_SWZ_F(mn, 2); mn = low_half ? fminf(mn, t) : fmaxf(mn, t);
      t = DS_SWZ_F(mn, 1); mn = low_half ? fminf(mn, t) : fmaxf(mn, t);
    }
    // lane 0 now holds the row min, lane 16 the row max.
    if ((lane == 0 && op == 0) || (lane == 16 && op != 0)) {
      out[row] = mn;
    }
  }
}

extern "C" void kernel_launch(void* const* d_in, const int* in_sizes, int n_in,
                              void* d_out, int out_size, void* d_ws, size_t ws_size,
                              hipStream_t stream) {
  const float* x    = (const float*)d_in[0];
  const int*   conn = (const int*)d_in[1];
  const int*   ops  = (const int*)d_in[2];
  float*       out  = (float*)d_out;

  deep_aggregate_kernel<<<NBLOCKS, THREADS, IN_FEATURES * sizeof(float), stream>>>(
      x, conn, ops, out);
}